// RelationGCN_24086176596515
// MI455X (gfx1250) — compile-verified
//
#include <hip/hip_runtime.h>
#include <hip/hip_bf16.h>

// RelationGCN for MI455X (gfx1250, wave32).
// Dense x@W via V_WMMA_F32_16X16X4_F32; sparse aggregation via float4 gather +
// global_atomic_add_f32; BN stats via LDS ds_add_f32 + global atomics.

#define GN 50000
#define GD 64
#define GE 800000

typedef float v2f __attribute__((ext_vector_type(2)));
typedef float v8f __attribute__((ext_vector_type(8)));

// ---------------- utility kernels ----------------

__global__ void k_fill(float* p, float v, int n) {
    int i = blockIdx.x * blockDim.x + threadIdx.x;
    if (i < n) p[i] = v;
}

__global__ void k_copy(const float* __restrict__ a, float* __restrict__ b, int n) {
    int i = blockIdx.x * blockDim.x + threadIdx.x;
    if (i < n) b[i] = a[i];
}

__global__ void k_deg_scatter(const int* __restrict__ dst, float* __restrict__ deg, int e) {
    int i = blockIdx.x * blockDim.x + threadIdx.x;
    if (i < e) atomicAdd(&deg[dst[i]], 1.0f);
}

__global__ void k_deg_rsqrt(float* deg, int n) {
    int i = blockIdx.x * blockDim.x + threadIdx.x;
    if (i < n) deg[i] = rsqrtf(deg[i]);   // deg >= 1 (self loop), matches deg^-0.5
}

// ---------------- dense: tmp = dinv[row] * ((emb * rel) @ W) ----------------
// One wave -> one 16x16 tile. 4 waves/block cover the 64 output columns.
// A (16x4 f32, 2 VGPR): lanes 0-15 M=0..15 K={0,1}; lanes 16-31 M=0..15 K={2,3}
// B (4x16 f32, 2 VGPR): lanes 0-15 N=0..15 K={0,1}; lanes 16-31 N=0..15 K={2,3}
// D (16x16 f32, 8 VGPR): VGPR v -> row v + 8*(lane>>4), col = lane&15

__global__ __launch_bounds__(128)
void k_gcn_matmul_wmma(const float* __restrict__ emb,
                       const float* __restrict__ rel,
                       const float* __restrict__ W,      // [64,64] row-major
                       const float* __restrict__ dinv,
                       float* __restrict__ tmp) {
    const int lane  = threadIdx.x & 31;
    const int wave  = threadIdx.x >> 5;
    const int r0    = blockIdx.x * 16;        // N = 50000 = 3125*16 exactly
    const int c0    = wave * 16;
    const int mrow  = lane & 15;              // M (for A) / N column (for B/D)
    const int khalf = lane >> 4;              // selects K pair {0,1} vs {2,3}

    const float* xrow = emb + (size_t)(r0 + mrow) * GD;

    v8f acc = {};
#pragma unroll
    for (int kk = 0; kk < 16; ++kk) {
        const int kb = kk * 4 + khalf * 2;
        // A fragment: two consecutive K values of this row, column-scaled by rel
        v2f a  = *(const v2f*)(xrow + kb);
        v2f rv = *(const v2f*)(rel + kb);
        a.x *= rv.x;
        a.y *= rv.y;
        // B fragment: W rows kb, kb+1 at column c0+mrow
        v2f b;
        b.x = W[(size_t)kb * GD + c0 + mrow];
        b.y = W[(size_t)(kb + 1) * GD + c0 + mrow];
        acc = __builtin_amdgcn_wmma_f32_16x16x4_f32(
            /*neg_a=*/false, a, /*neg_b=*/false, b,
            /*c_mod=*/(short)0, acc, /*reuse_a=*/false, /*reuse_b=*/false);
    }

#pragma unroll
    for (int v = 0; v < 8; ++v) {
        const int row = r0 + v + 8 * khalf;
        tmp[(size_t)row * GD + c0 + mrow] = acc[v] * dinv[row];
    }
}

// ---------------- sparse: agg[d] += tmp[s] (plus self loops) ----------------
// 16 threads per edge, float4 per thread -> coalesced 256B row gathers.

__global__ void k_gcn_scatter(const float* __restrict__ tmp,
                              const int* __restrict__ src,
                              const int* __restrict__ dst,
                              float* __restrict__ agg) {
    const int gid = blockIdx.x * blockDim.x + threadIdx.x;
    const int total = (GE + GN) * 16;
    if (gid >= total) return;
    const int e = gid >> 4;
    const int f = (gid & 15) * 4;
    int s, d;
    if (e < GE) { s = src[e]; d = dst[e]; }
    else        { s = e - GE; d = s; }          // self loop
    const float4 v = *(const float4*)(tmp + (size_t)s * GD + f);
    float* ap = agg + (size_t)d * GD + f;
    atomicAdd(ap + 0, v.x);
    atomicAdd(ap + 1, v.y);
    atomicAdd(ap + 2, v.z);
    atomicAdd(ap + 3, v.w);
}

// ---------------- BN statistics over h = dinv[r]*agg[r][c] + b[c] ----------------

__global__ __launch_bounds__(256)
void k_bn_stats(const float* __restrict__ agg,
                const float* __restrict__ dinv,
                const float* __restrict__ bias,
                float* __restrict__ stats) {      // [0:64) sum, [64:128) sumsq
    __shared__ float s_sum[GD];
    __shared__ float s_sq[GD];
    const int tid = threadIdx.x;
    if (tid < GD) { s_sum[tid] = 0.0f; s_sq[tid] = 0.0f; }
    __syncthreads();

    const int c  = tid & 63;
    const int rb = blockIdx.x * 4 + (tid >> 6);
    const float bc = bias[c];
    float lsum = 0.0f, lsq = 0.0f;
    for (int r = rb; r < GN; r += gridDim.x * 4) {
        const float h = dinv[r] * agg[(size_t)r * GD + c] + bc;
        lsum += h;
        lsq  += h * h;
    }
    atomicAdd(&s_sum[c], lsum);   // LDS float atomic (ds_add_f32)
    atomicAdd(&s_sq[c], lsq);
    __syncthreads();
    if (tid < GD) {
        atomicAdd(&stats[tid], s_sum[tid]);
        atomicAdd(&stats[GD + tid], s_sq[tid]);
    }
}

__global__ void k_bn_finalize(float* __restrict__ stats,
                              const float* __restrict__ gamma,
                              const float* __restrict__ beta) {
    const int c = threadIdx.x;                 // 64 threads
    const float invN = 1.0f / (float)GN;
    const float mean = stats[c] * invN;
    const float var  = stats[GD + c] * invN - mean * mean;
    const float inv  = rsqrtf(var + 1e-5f);
    const float sc   = gamma[c] * inv;
    stats[128 + c] = sc;                       // scale
    stats[192 + c] = beta[c] - mean * sc;      // shift
}

// emb += leaky_relu(scale*(dinv*agg + b) + shift)
__global__ void k_emb_update(float* __restrict__ emb,
                             const float* __restrict__ agg,
                             const float* __restrict__ dinv,
                             const float* __restrict__ bias,
                             const float* __restrict__ stats) {
    const int gid = blockIdx.x * blockDim.x + threadIdx.x;
    if (gid >= GN * GD) return;
    const int c = gid & 63;
    const int r = gid >> 6;
    const float h = dinv[r] * agg[gid] + bias[c];
    const float v = stats[128 + c] * h + stats[192 + c];
    emb[gid] += (v >= 0.0f) ? v : 0.01f * v;
}

// final layer: emb = dinv*agg + b   (no BN / residual)
__global__ void k_emb_final(float* __restrict__ emb,
                            const float* __restrict__ agg,
                            const float* __restrict__ dinv,
                            const float* __restrict__ bias) {
    const int gid = blockIdx.x * blockDim.x + threadIdx.x;
    if (gid >= GN * GD) return;
    const int c = gid & 63;
    const int r = gid >> 6;
    emb[gid] = dinv[r] * agg[gid] + bias[c];
}

// rel = rel @ Wr^T + br  -> out[j] = sum_k rel[k]*Wr[j*64+k] + br[j]; single block.
__global__ void k_rel_update(float* __restrict__ rel,
                             const float* __restrict__ Wr,
                             const float* __restrict__ br) {
    __shared__ float oldr[GD];
    const int j = threadIdx.x;                 // 64 threads
    oldr[j] = rel[j];
    __syncthreads();
    float acc = br[j];
#pragma unroll 8
    for (int k = 0; k < GD; ++k) acc += oldr[k] * Wr[(size_t)j * GD + k];
    rel[j] = acc;
}

// ---------------- host orchestration ----------------

extern "C" void kernel_launch(void* const* d_in, const int* in_sizes, int n_in,
                              void* d_out, int out_size, void* d_ws, size_t ws_size,
                              hipStream_t stream) {
    (void)in_sizes; (void)n_in; (void)out_size; (void)ws_size;

    const float* features = (const float*)d_in[0];
    // Relation order in outputs: n, poi, s, d. Inputs: poi_r(1) s_r(2) d_r(3) n_r(4),
    // poi_edge(5) s_edge(6) d_edge(7) n_edge(8).
    const float* rel_in[4]  = { (const float*)d_in[4], (const float*)d_in[1],
                                (const float*)d_in[2], (const float*)d_in[3] };
    const int*   edge_in[4] = { (const int*)d_in[8], (const int*)d_in[5],
                                (const int*)d_in[6], (const int*)d_in[7] };
    const float* gcn_W    = (const float*)d_in[9];   // [3,64,64]
    const float* gcn_b    = (const float*)d_in[10];  // [3,64]
    const float* bn_gamma = (const float*)d_in[11];  // [2,64]
    const float* bn_beta  = (const float*)d_in[12];  // [2,64]
    const float* rel_W    = (const float*)d_in[13];  // [3,64,64]
    const float* rel_b    = (const float*)d_in[14];  // [3,64]

    float* out = (float*)d_out;
    float* emb[4];
    float* rel[4];
    for (int k = 0; k < 4; ++k) {
        emb[k] = out + (size_t)k * GN * GD;
        rel[k] = out + (size_t)4 * GN * GD + (size_t)k * GD;
    }

    // Workspace: tmp[N*64] | agg[N*64] | dinv[4*N] | stats[256]
    float* tmp   = (float*)d_ws;
    float* agg   = tmp + (size_t)GN * GD;
    float* dinvb = agg + (size_t)GN * GD;
    float* stats = dinvb + (size_t)4 * GN;

    const int TB = 256;
    const int gN    = (GN + TB - 1) / TB;
    const int gE    = (GE + TB - 1) / TB;
    const int gND   = (GN * GD + TB - 1) / TB;
    const int gScat = ((GE + GN) * 16 + TB - 1) / TB;

    // Degrees (structure only; computed once). Self loop => start at 1.0.
    for (int k = 0; k < 4; ++k)
        k_fill<<<gN, TB, 0, stream>>>(dinvb + (size_t)k * GN, 1.0f, GN);
    for (int k = 0; k < 4; ++k)
        k_deg_scatter<<<gE, TB, 0, stream>>>(edge_in[k] + GE /*dst row*/,
                                             dinvb + (size_t)k * GN, GE);
    for (int k = 0; k < 4; ++k)
        k_deg_rsqrt<<<gN, TB, 0, stream>>>(dinvb + (size_t)k * GN, GN);

    // Initialize persistent state in d_out.
    for (int k = 0; k < 4; ++k) {
        k_copy<<<gND, TB, 0, stream>>>(features, emb[k], GN * GD);
        k_copy<<<1, GD, 0, stream>>>(rel_in[k], rel[k], GD);
    }

    for (int layer = 0; layer < 3; ++layer) {
        const float* Wl  = gcn_W + (size_t)layer * GD * GD;
        const float* bl  = gcn_b + (size_t)layer * GD;
        const float* Wrl = rel_W + (size_t)layer * GD * GD;
        const float* brl = rel_b + (size_t)layer * GD;

        for (int k = 0; k < 4; ++k) {
            const float* dinv = dinvb + (size_t)k * GN;
            const int* src = edge_in[k];
            const int* dst = edge_in[k] + GE;

            k_fill<<<gND, TB, 0, stream>>>(agg, 0.0f, GN * GD);
            k_gcn_matmul_wmma<<<GN / 16, 128, 0, stream>>>(emb[k], rel[k], Wl, dinv, tmp);
            k_gcn_scatter<<<gScat, TB, 0, stream>>>(tmp, src, dst, agg);

            if (layer < 2) {
                k_fill<<<1, 128, 0, stream>>>(stats, 0.0f, 128);
                k_bn_stats<<<512, 256, 0, stream>>>(agg, dinv, bl, stats);
                k_bn_finalize<<<1, GD, 0, stream>>>(stats,
                                                    bn_gamma + (size_t)layer * GD,
                                                    bn_beta + (size_t)layer * GD);
                k_emb_update<<<gND, TB, 0, stream>>>(emb[k], agg, dinv, bl, stats);
            } else {
                k_emb_final<<<gND, TB, 0, stream>>>(emb[k], agg, dinv, bl);
            }
            k_rel_update<<<1, GD, 0, stream>>>(rel[k], Wrl, brl);
        }
    }
}